// ParallelActions_32100585570782
// MI455X (gfx1250) — compile-verified
//
#include <hip/hip_runtime.h>

typedef __attribute__((ext_vector_type(2))) float v2f;
typedef __attribute__((ext_vector_type(8))) float v8f;

#define TDIM 16
#define NACT 8

// ---------------------------------------------------------------------------
// Kernel 1: W[a] = inv(A[a]) @ B_rep[a] @ A[a], one wave per action.
// Gauss-Jordan with partial pivoting on [A | I] in LDS, lanes parallel over
// the 32 columns of the augmented matrix. Output row-major W[8][16][16].
// ---------------------------------------------------------------------------
__global__ __launch_bounds__(256) void prep_weights(const float* __restrict__ Ag,
                                                    const float* __restrict__ Bg,
                                                    float* __restrict__ Wg)
{
    __shared__ float aug[NACT][TDIM][2 * TDIM];
    __shared__ float tmp[NACT][TDIM][TDIM];
    const int w    = threadIdx.x >> 5;   // wave id == action id
    const int lane = threadIdx.x & 31;
    const float* A = Ag + w * TDIM * TDIM;
    const float* B = Bg + w * TDIM * TDIM;

    // augmented [A | I]
    for (int r = 0; r < TDIM; ++r) {
        float v;
        if (lane < TDIM) v = A[r * TDIM + lane];
        else             v = (lane - TDIM == r) ? 1.0f : 0.0f;
        aug[w][r][lane] = v;
    }
    __syncthreads();

    for (int k = 0; k < TDIM; ++k) {
        // partial pivot search: redundant across lanes, uniform result
        int prow = k;
        float best = fabsf(aug[w][k][k]);
        for (int r = k + 1; r < TDIM; ++r) {
            float v = fabsf(aug[w][r][k]);
            if (v > best) { best = v; prow = r; }
        }
        if (prow != k) {                 // uniform branch; lane swaps its column
            float t0 = aug[w][k][lane];
            float t1 = aug[w][prow][lane];
            aug[w][k][lane]    = t1;
            aug[w][prow][lane] = t0;
        }
        __syncthreads();
        float piv = aug[w][k][k];
        __syncthreads();
        aug[w][k][lane] = aug[w][k][lane] / piv;
        __syncthreads();
        float rowk = aug[w][k][lane];
        for (int r = 0; r < TDIM; ++r) {
            if (r == k) continue;
            float f = aug[w][r][k];      // broadcast read before lane k writes
            aug[w][r][lane] = aug[w][r][lane] - f * rowk;
        }
        __syncthreads();
    }

    // tmp = invA @ B   (invA[i][k] lives in aug[w][i][16+k])
    for (int t = 0; t < 8; ++t) {
        int c = lane + 32 * t;           // 0..255
        int i = c >> 4, j = c & 15;
        float s = 0.f;
        for (int k2 = 0; k2 < TDIM; ++k2)
            s += aug[w][i][TDIM + k2] * B[k2 * TDIM + j];
        tmp[w][i][j] = s;
    }
    __syncthreads();

    // W = tmp @ A  -> global scratch, row-major
    for (int t = 0; t < 8; ++t) {
        int c = lane + 32 * t;
        int i = c >> 4, j = c & 15;
        float s = 0.f;
        for (int k2 = 0; k2 < TDIM; ++k2)
            s += tmp[w][i][k2] * A[k2 * TDIM + j];
        Wg[w * TDIM * TDIM + c] = s;
    }
}

// ---------------------------------------------------------------------------
// Kernel 2: K-expanded GEMM per 16-sample tile.
//   Y[s,i] = sum_{k=0..127} XE[s,k] * Bcat[k,i]
//   XE[s, a*16+j] = (a == act_s) ? x[s,j] : 0 ;  Bcat[a*16+j, i] = W_a[i,j]
// Selection is done by zeroing the A operand with the lane's OWN action
// (no cross-lane shuffle), so a single pair of accumulators suffices.
// 32 x V_WMMA_F32_16X16X4_F32 per tile; B operands resident in 64 VGPRs.
// ---------------------------------------------------------------------------
template<bool FULL>
__device__ __forceinline__ void do_tile(const float* __restrict__ x,
                                        const int*   __restrict__ act,
                                        float*       __restrict__ y,
                                        const v2f (&bv)[NACT][4],
                                        int tb, int n, int r, int hi, int lane)
{
    const int row = tb + r;

    v2f xv[4];
    int myAct;
    if (FULL) {
        #pragma unroll
        for (int ks = 0; ks < 4; ++ks)
            xv[ks] = *(const v2f*)(x + row * 16 + 4 * ks + 2 * hi);
        myAct = act[row];
    } else {
        const bool rok = row < n;
        #pragma unroll
        for (int ks = 0; ks < 4; ++ks) {
            v2f v = {0.f, 0.f};
            if (rok) v = *(const v2f*)(x + row * 16 + 4 * ks + 2 * hi);
            xv[ks] = v;
        }
        myAct = rok ? act[row] : -1;
    }

    // two independent accumulation chains (even/odd actions) for pipelining
    v8f acc0 = {0.f, 0.f, 0.f, 0.f, 0.f, 0.f, 0.f, 0.f};
    v8f acc1 = {0.f, 0.f, 0.f, 0.f, 0.f, 0.f, 0.f, 0.f};
    #pragma unroll
    for (int a = 0; a < NACT; a += 2) {
        const bool m0 = (myAct == a);
        const bool m1 = (myAct == a + 1);
        #pragma unroll
        for (int ks = 0; ks < 4; ++ks) {
            v2f xa0, xa1;
            xa0.x = m0 ? xv[ks].x : 0.f;
            xa0.y = m0 ? xv[ks].y : 0.f;
            xa1.x = m1 ? xv[ks].x : 0.f;
            xa1.y = m1 ? xv[ks].y : 0.f;
            acc0 = __builtin_amdgcn_wmma_f32_16x16x4_f32(
                       false, xa0, false, bv[a][ks],     (short)0, acc0, false, false);
            acc1 = __builtin_amdgcn_wmma_f32_16x16x4_f32(
                       false, xa1, false, bv[a + 1][ks], (short)0, acc1, false, false);
        }
    }

    // D layout: VGPR v on lane holds y[tb + v + 8*hi][lane&15]
    #pragma unroll
    for (int v = 0; v < 8; ++v) {
        const int s = tb + v + 8 * hi;
        const float val = acc0[v] + acc1[v];
        if (FULL) {
            y[s * 16 + (lane & 15)] = val;
        } else if (s < n) {
            y[s * 16 + (lane & 15)] = val;
        }
    }
}

__global__ __launch_bounds__(256) void gemv_select(const float* __restrict__ x,
                                                   const int*   __restrict__ act,
                                                   const float* __restrict__ Wg,
                                                   float*       __restrict__ y,
                                                   int n, int ntiles)
{
    const int lane        = threadIdx.x & 31;
    const int waveInBlk   = threadIdx.x >> 5;
    const int wavesPerBlk = blockDim.x >> 5;
    const int gwave  = blockIdx.x * wavesPerBlk + waveInBlk;
    const int nwaves = gridDim.x * wavesPerBlk;

    const int r  = lane & 15;            // row within half-wave
    const int hi = lane >> 4;            // 0 or 1

    // B operand regs: Bcat[a*16+j][i] = W_a[i][j]; slice pair = W[a][r][4ks+2hi..+1]
    v2f bv[NACT][4];
    #pragma unroll
    for (int a = 0; a < NACT; ++a) {
        #pragma unroll
        for (int ks = 0; ks < 4; ++ks) {
            const float* p = Wg + a * 256 + r * 16 + 4 * ks + 2 * hi;
            bv[a][ks] = *(const v2f*)p;
        }
    }

    for (int t = gwave; t < ntiles; t += nwaves) {
        const int tb = t * 16;
        if (tb + 16 <= n) {              // wave-uniform fast path (almost always)
            do_tile<true>(x, act, y, bv, tb, n, r, hi, lane);
        } else {
            do_tile<false>(x, act, y, bv, tb, n, r, hi, lane);
        }
    }
}

// ---------------------------------------------------------------------------
extern "C" void kernel_launch(void* const* d_in, const int* in_sizes, int n_in,
                              void* d_out, int out_size, void* d_ws, size_t ws_size,
                              hipStream_t stream) {
    const float* x    = (const float*)d_in[0];
    const int*   a    = (const int*)  d_in[1];
    const float* A    = (const float*)d_in[2];
    const float* Brep = (const float*)d_in[3];
    float* y  = (float*)d_out;
    float* Wg = (float*)d_ws;              // 8*16*16 floats = 8 KB

    const int n = in_sizes[1];             // batch size (actions array length)

    prep_weights<<<1, 256, 0, stream>>>(A, Brep, Wg);

    const int ntiles = (n + 15) / 16;
    const int wavesPerBlk  = 8;            // 256 threads = 8 wave32
    const int tilesPerWave = 4;            // amortize B-operand loads
    int blocks = (ntiles + wavesPerBlk * tilesPerWave - 1) / (wavesPerBlk * tilesPerWave);
    if (blocks < 1) blocks = 1;
    if (blocks > 32768) blocks = 32768;
    gemv_select<<<blocks, 256, 0, stream>>>(x, a, Wg, y, n, ntiles);
}